// Custom_Model_Embedding_Bag_Sum_nodes_2834678415999
// MI455X (gfx1250) — compile-verified
//
#include <hip/hip_runtime.h>
#include <hip/hip_bf16.h>

// CDNA5 / gfx1250. Embedding-bag + global sum collapses to:
//   out = per-table sums of W[t, eb_input[i], :] over ALL 4M indices,
//   with tables 5,6 additionally reduced over the 3-dim axis. 26 floats out.
// Memory-bound random gather (120MB table set fits 192MB L2). Main loop is
// per-thread register accumulation with 10 independent b96 gathers/index;
// block reduction is done on the matrix pipe: colsums = ones^T x S via
// chained V_WMMA_F32_16X16X4_F32 (A = ones -> K-order-invariant column sums).

typedef float v2f __attribute__((ext_vector_type(2)));
typedef float v8f __attribute__((ext_vector_type(8)));

#define N_TABLES     10
#define EMB_DIM      3
#define VOCAB_SZ     1000000
#define TABLE_STRIDE (VOCAB_SZ * EMB_DIM)   // floats per table
#define NCOMP        30                     // 10 tables x 3 dims
#define LDS_STRIDE   33                     // pad 32+1 -> conflict-free rows
#define BLOCK        256

__global__ void zero_out_kernel(float* __restrict__ out, int n) {
  int i = blockIdx.x * blockDim.x + threadIdx.x;
  if (i < n) out[i] = 0.0f;
}

struct F3 { float x, y, z; };  // 12-byte row -> global_load_b96

__global__ __launch_bounds__(BLOCK)
void embbag_sum_kernel(const int* __restrict__ eb_input,
                       const float* __restrict__ W,
                       float* __restrict__ out,
                       int n_idx) {
  // ---- per-thread register accumulators (30 floats) ----
  float acc[NCOMP];
#pragma unroll
  for (int c = 0; c < NCOMP; ++c) acc[c] = 0.0f;

  const int tid    = blockIdx.x * blockDim.x + threadIdx.x;
  const int stride = gridDim.x * blockDim.x;

  // Software-pipelined grid-stride loop: next index load overlaps the 10
  // independent row gathers for the current index.
  int i  = tid;
  int id = (i < n_idx) ? eb_input[i] : 0;
  while (i < n_idx) {
    const int inext = i + stride;
    // prefetch the index stream two strides ahead (global_prefetch_b8;
    // speculative -> OOB addresses are silently dropped)
    __builtin_prefetch(&eb_input[i + 2 * stride], 0, 0);
    const int idnext = (inext < n_idx) ? eb_input[inext] : 0;

    const float* row = W + (size_t)id * EMB_DIM;
#pragma unroll
    for (int t = 0; t < N_TABLES; ++t) {
      F3 v = *reinterpret_cast<const F3*>(row + (size_t)t * TABLE_STRIDE);
      acc[t * 3 + 0] += v.x;
      acc[t * 3 + 1] += v.y;
      acc[t * 3 + 2] += v.z;
    }
    id = idnext;
    i  = inext;
  }

  // ---- stage per-thread partials into LDS: S[256][33] (stride-33 pad) ----
  __shared__ float S[BLOCK * LDS_STRIDE];
  {
    const int r = threadIdx.x;
#pragma unroll
    for (int c = 0; c < NCOMP; ++c) S[r * LDS_STRIDE + c] = acc[c];
    S[r * LDS_STRIDE + 30] = 0.0f;   // pad columns feeding unused WMMA cols
    S[r * LDS_STRIDE + 31] = 0.0f;
  }
  __syncthreads();

  // ---- WMMA reduction: colsums(32) = ones(16x4) x S-chunks(4x16), chained C.
  // Wave 0 only (EXEC must be all ones per-wave for WMMA).
  if (threadIdx.x < 32) {
    const int lane = threadIdx.x;
    const int h    = lane >> 4;    // which 16-lane half
    const int cc   = lane & 15;    // column within half

    v2f ones; ones[0] = 1.0f; ones[1] = 1.0f;      // A = ones(16x4)
    v8f dA = {0.f, 0.f, 0.f, 0.f, 0.f, 0.f, 0.f, 0.f};  // comps 0..15
    v8f dB = {0.f, 0.f, 0.f, 0.f, 0.f, 0.f, 0.f, 0.f};  // comps 16..31

    for (int rr = 0; rr < BLOCK; rr += 4) {
      // Column cc of B gathers rows rr..rr+3 across {v0,v1} x {lane halves};
      // with A==ones the K-slot ordering is irrelevant.
      const int ra = (rr + h) * LDS_STRIDE;
      const int rb = (rr + 2 + h) * LDS_STRIDE;
      v2f bA, bB;
      bA[0] = S[ra + cc];        bA[1] = S[rb + cc];
      bB[0] = S[ra + 16 + cc];   bB[1] = S[rb + 16 + cc];
      dA = __builtin_amdgcn_wmma_f32_16x16x4_f32(
               false, ones, false, bA, (short)0, dA, false, false);
      dB = __builtin_amdgcn_wmma_f32_16x16x4_f32(
               false, ones, false, bB, (short)0, dB, false, false);
    }

    // D VGPR0: lanes 0-15 -> (M=0, N=lane), lanes 16-31 -> (M=8, N=lane-16);
    // every row of D equals the column sums, so element 0 per lane suffices.
    // Component index c == lane for both halves (half 1 holds comps 16..31).
    const float colsum = (lane < 16) ? dA[0] : dB[0];
    const int c = lane;
    if (c < NCOMP) {
      // Map component -> output slot, folding tables 5 and 6 to scalars:
      // out[0..14]=t0..t4, out[15]=sum(t5), out[16]=sum(t6), out[17..25]=t7..t9
      const int m = (c < 15) ? c : (c < 18 ? 15 : (c < 21 ? 16 : c - 4));
      atomicAdd(&out[m], colsum);
    }
  }
}

extern "C" void kernel_launch(void* const* d_in, const int* in_sizes, int n_in,
                              void* d_out, int out_size, void* d_ws, size_t ws_size,
                              hipStream_t stream) {
  const int*   eb_input = (const int*)d_in[0];
  // d_in[1] = eb_offset : unused — summing over all bags collapses segments.
  const float* W        = (const float*)d_in[2];
  float*       out      = (float*)d_out;
  const int    n_idx    = in_sizes[0];

  zero_out_kernel<<<1, 32, 0, stream>>>(out, out_size);

  // ~164K threads -> ~24 indices/thread; 30 float atomics per block total.
  const int blocks = 640;
  embbag_sum_kernel<<<blocks, BLOCK, 0, stream>>>(eb_input, W, out, n_idx);
}